// ENCELDDT_67602785239182
// MI455X (gfx1250) — compile-verified
//
#include <hip/hip_runtime.h>
#include <math.h>

#define NPTS 4096
#define NUM_BINS 10
#define TILE 16
#define TPD (NPTS / TILE)          // 256 tiles per dimension
#define TOTAL_TILES (TPD * TPD)    // 65536 tiles
#define WAVES_PER_BLOCK 8

typedef __attribute__((ext_vector_type(2))) float v2f;
typedef __attribute__((ext_vector_type(8))) float v8f;

// ---------------------------------------------------------------------------
// Kernel 1: two smallest / two largest sigmas -> begin & interval (sig2 is
// monotone decreasing in sigma_i+sigma_j). Precise math here (runs once).
// Also zeroes the three accumulators.
// ---------------------------------------------------------------------------
__global__ __launch_bounds__(1024) void ENC_minmax_init(
    const float* __restrict__ sigmas, float* __restrict__ hdr)
{
    __shared__ float sm[2048];   // two smallest per thread
    __shared__ float sM[2048];   // two largest per thread
    const int t = threadIdx.x;

    float a1 = 3.4e38f, a2 = 3.4e38f;
    float b1 = -3.4e38f, b2 = -3.4e38f;
    for (int i = t; i < NPTS; i += 1024) {
        float v = sigmas[i];
        if (v < a1) { a2 = a1; a1 = v; } else if (v < a2) { a2 = v; }
        if (v > b1) { b2 = b1; b1 = v; } else if (v > b2) { b2 = v; }
    }
    sm[2*t] = a1; sm[2*t+1] = a2;
    sM[2*t] = b1; sM[2*t+1] = b2;
    __syncthreads();

    for (int s = 512; s > 0; s >>= 1) {
        if (t < s) {
            float m1 = sm[2*t], m2 = sm[2*t+1];
            float o1 = sm[2*(t+s)], o2 = sm[2*(t+s)+1];
            if (o1 < m1) { m2 = m1; m1 = o1; } else if (o1 < m2) { m2 = o1; }
            if (o2 < m1) { m2 = m1; m1 = o2; } else if (o2 < m2) { m2 = o2; }
            sm[2*t] = m1; sm[2*t+1] = m2;

            float M1 = sM[2*t], M2 = sM[2*t+1];
            float p1 = sM[2*(t+s)], p2 = sM[2*(t+s)+1];
            if (p1 > M1) { M2 = M1; M1 = p1; } else if (p1 > M2) { M2 = p1; }
            if (p2 > M1) { M2 = M1; M1 = p2; } else if (p2 > M2) { M2 = p2; }
            sM[2*t] = M1; sM[2*t+1] = M2;
        }
        __syncthreads();
    }

    if (t == 0) {
        float avg_hi = (sM[0] + sM[1]) / 200.0f;   // largest avg -> smallest sig2
        float avg_lo = (sm[0] + sm[1]) / 200.0f;   // smallest avg -> largest sig2
        float begin = -2.0f / logf(1.0f - avg_hi * avg_hi);
        float end   = -2.0f / logf(1.0f - avg_lo * avg_lo);
        hdr[0] = begin;
        hdr[1] = (end - begin) / (float)NUM_BINS;  // interval
        hdr[2] = 0.0f;  // cnt
        hdr[3] = 0.0f;  // sum sig2
        hdr[4] = 0.0f;  // sum tr2
    }
}

// ---------------------------------------------------------------------------
// Kernel 2: per-row squared norms of y and py.
// ---------------------------------------------------------------------------
__global__ __launch_bounds__(256) void ENC_norms(
    const float* __restrict__ Y, const float* __restrict__ PY,
    float* __restrict__ ny, float* __restrict__ np)
{
    int i = blockIdx.x * 256 + threadIdx.x;
    if (i < NPTS) {
        float a = Y[3*i], b = Y[3*i+1], c = Y[3*i+2];
        ny[i] = a*a + b*b + c*c;
        float d = PY[3*i], e = PY[3*i+1], f = PY[3*i+2];
        np[i] = d*d + e*e + f*f;
    }
}

// ---------------------------------------------------------------------------
// Kernel 3: main pass. One wave32 per 16x16 tile of the pair matrix.
// Gram matrices via V_WMMA_F32_16X16X4_F32 (K padded 3->4 with zeros),
// then a branchless fast-math epilogue (raw v_sqrt/v_log/v_rcp — args are
// never denormal/out-of-range here) + wave butterfly reduction.
// ---------------------------------------------------------------------------
__global__ __launch_bounds__(256) void ENC_pair_tiles(
    const float* __restrict__ sigmas,
    const float* __restrict__ Y,
    const float* __restrict__ PY,
    const float* __restrict__ ny,
    const float* __restrict__ np,
    const float* __restrict__ hdr,
    float* __restrict__ acc)
{
    const int lane = threadIdx.x & 31;
    const int wave = threadIdx.x >> 5;
    const int tile = blockIdx.x * WAVES_PER_BLOCK + wave;
    const int bi = tile >> 8;
    const int bj = tile & (TPD - 1);

    const int m  = lane & 15;
    const bool hi = (lane >= 16);
    const int gi = bi * TILE + m;
    const int gj = bj * TILE + m;

    // A (16x4 f32): lanes 0-15 = (K0,K1) of row M=lane; lanes 16-31 = (K2,0).
    // B (4x16 f32): same striping with N across lanes.
    v2f a_y, b_y, a_p, b_p;
    if (!hi) {
        a_y = (v2f){ Y[3*gi],  Y[3*gi+1]  };
        b_y = (v2f){ Y[3*gj],  Y[3*gj+1]  };
        a_p = (v2f){ PY[3*gi], PY[3*gi+1] };
        b_p = (v2f){ PY[3*gj], PY[3*gj+1] };
    } else {
        a_y = (v2f){ Y[3*gi+2],  0.0f };
        b_y = (v2f){ Y[3*gj+2],  0.0f };
        a_p = (v2f){ PY[3*gi+2], 0.0f };
        b_p = (v2f){ PY[3*gj+2], 0.0f };
    }

    v8f gy = {};
    v8f gp = {};
    gy = __builtin_amdgcn_wmma_f32_16x16x4_f32(false, a_y, false, b_y,
                                               (short)0, gy, false, false);
    gp = __builtin_amdgcn_wmma_f32_16x16x4_f32(false, a_p, false, b_p,
                                               (short)0, gp, false, false);

    const float begin = hdr[0];
    const float invI  = __builtin_amdgcn_rcpf(hdr[1]);  // once per wave

    const int j   = bj * TILE + m;      // column handled by this lane
    const float sj  = sigmas[j];
    const float nyj = ny[j];
    const float npj = np[j];
    const int mbase = hi ? 8 : 0;

    float c_cnt = 0.0f, c_ss = 0.0f, c_ts = 0.0f;
    #pragma unroll
    for (int r = 0; r < 8; ++r) {
        const int i = bi * TILE + mbase + r;
        float nyi = ny[i], npi = np[i], si = sigmas[i];
        // dist^2 = |yi|^2 + |yj|^2 - 2*Gram  (clamped at 0)
        float d2g = fmaxf(fmaf(-2.0f, gy[r], nyi + nyj), 0.0f);
        float d2p = fmaxf(fmaf(-2.0f, gp[r], npi + npj), 0.0f);
        float tr  = __builtin_amdgcn_sqrtf(d2g) - __builtin_amdgcn_sqrtf(d2p);
        float avg = (si + sj) * (1.0f / 200.0f);
        float L   = __logf(fmaf(-avg, avg, 1.0f));       // in (0.02, 1): no fixup needed
        float s2  = -2.0f * __builtin_amdgcn_rcpf(L);
        float t   = (s2 - begin) * invI;                 // bin = clip(floor(t),0,9)
        bool keep = (t >= (float)(NUM_BINS - 1)) && (i != j);
        c_cnt += keep ? 1.0f    : 0.0f;
        c_ss  += keep ? s2      : 0.0f;
        c_ts  += keep ? tr * tr : 0.0f;
    }

    // wave32 butterfly reduction
    #pragma unroll
    for (int off = 16; off >= 1; off >>= 1) {
        c_cnt += __shfl_xor(c_cnt, off, 32);
        c_ss  += __shfl_xor(c_ss,  off, 32);
        c_ts  += __shfl_xor(c_ts,  off, 32);
    }
    if (lane == 0) {
        atomicAdd(&acc[0], c_cnt);
        atomicAdd(&acc[1], c_ss);
        atomicAdd(&acc[2], c_ts);
    }
}

// ---------------------------------------------------------------------------
// Kernel 4: finalize -> scalar output (precise math, runs once).
// ---------------------------------------------------------------------------
__global__ void ENC_finalize(const float* __restrict__ acc, float* __restrict__ out)
{
    float cnt = acc[0];
    float mvar = sqrtf(acc[1] / cnt);
    float rmse = sqrtf(acc[2] / cnt);
    out[0] = fabsf(mvar - rmse) / mvar;
}

extern "C" void kernel_launch(void* const* d_in, const int* in_sizes, int n_in,
                              void* d_out, int out_size, void* d_ws, size_t ws_size,
                              hipStream_t stream) {
    const float* sigmas = (const float*)d_in[0];
    const float* y      = (const float*)d_in[1];
    const float* py     = (const float*)d_in[2];
    float* out = (float*)d_out;

    float* ws  = (float*)d_ws;
    float* hdr = ws;               // [0]=begin [1]=interval [2..4]=accumulators
    float* acc = ws + 2;
    float* ny  = ws + 8;           // 4096 floats
    float* np  = ws + 8 + NPTS;    // 4096 floats

    ENC_minmax_init<<<1, 1024, 0, stream>>>(sigmas, hdr);
    ENC_norms<<<(NPTS + 255) / 256, 256, 0, stream>>>(y, py, ny, np);
    ENC_pair_tiles<<<TOTAL_TILES / WAVES_PER_BLOCK, 256, 0, stream>>>(
        sigmas, y, py, ny, np, hdr, acc);
    ENC_finalize<<<1, 1, 0, stream>>>(acc, out);
}